// Blis_33139967655989
// MI455X (gfx1250) — compile-verified
//
#include <hip/hip_runtime.h>

// ---------------------------------------------------------------------------
// Blis graph-wavelet scattering for MI455X (gfx1250, wave32).
//  - 16 diffusion steps: scatter-add SpMM, state resident in 192MB L2,
//    native f32 atomics (global_atomic_add_f32), float4-coalesced lanes.
//  - Wavelet combine W[6,17] @ levels[17,N*C] done with V_WMMA_F32_16X16X4_F32
//    chunked K=4 over level groups (4-buffer ring), f32 accumulation via the
//    WMMA C operand round-tripped through a [6,N*C] coeff buffer.
//  - Final relu(+/-) expansion with contiguous float2 stores.
// ---------------------------------------------------------------------------

typedef __attribute__((ext_vector_type(2))) float v2f;
typedef __attribute__((ext_vector_type(8))) float v8f;

#define C_FEAT 128

__global__ void k_deg(const int* __restrict__ col, float* __restrict__ deg, int E) {
  int e = blockIdx.x * blockDim.x + threadIdx.x;
  if (e < E)
    __hip_atomic_fetch_add(&deg[col[e]], 1.0f, __ATOMIC_RELAXED, __HIP_MEMORY_SCOPE_AGENT);
}

__global__ void k_inv(float* __restrict__ deg, int n) {
  int i = blockIdx.x * blockDim.x + threadIdx.x;
  if (i < n) {
    float d = deg[i];
    deg[i] = (d > 0.f) ? (1.f / d) : 0.f;
  }
}

__global__ void k_gatherw(const int* __restrict__ col, const float* __restrict__ deginv,
                          float* __restrict__ w, int E) {
  int e = blockIdx.x * blockDim.x + threadIdx.x;
  if (e < E) w[e] = deginv[col[e]];
}

// One wave32 per edge: 32 lanes x float4 = all 128 features, coalesced 512B.
__global__ void k_scatter(const int* __restrict__ row, const int* __restrict__ col,
                          const float* __restrict__ w, const float* __restrict__ s,
                          float* __restrict__ prop, int E) {
  int e = blockIdx.x * (blockDim.x >> 5) + (threadIdx.x >> 5);
  if (e >= E) return;
  int c = (threadIdx.x & 31) << 2;
  long bs = (long)col[e] * C_FEAT + c;
  long bp = (long)row[e] * C_FEAT + c;
  float we = w[e];
  float4 sv = *reinterpret_cast<const float4*>(s + bs);
  float* p = prop + bp;
  __hip_atomic_fetch_add(p + 0, we * sv.x, __ATOMIC_RELAXED, __HIP_MEMORY_SCOPE_AGENT);
  __hip_atomic_fetch_add(p + 1, we * sv.y, __ATOMIC_RELAXED, __HIP_MEMORY_SCOPE_AGENT);
  __hip_atomic_fetch_add(p + 2, we * sv.z, __ATOMIC_RELAXED, __HIP_MEMORY_SCOPE_AGENT);
  __hip_atomic_fetch_add(p + 3, we * sv.w, __ATOMIC_RELAXED, __HIP_MEMORY_SCOPE_AGENT);
}

// s_new = 0.5*(s_prev + prop), written in place over prop.
__global__ void k_combine(const float* __restrict__ sprev, float* __restrict__ sp, long n4) {
  long i = (long)blockIdx.x * blockDim.x + threadIdx.x;
  if (i >= n4) return;
  float4 a = reinterpret_cast<const float4*>(sprev)[i];
  float4 b = reinterpret_cast<float4*>(sp)[i];
  b.x = 0.5f * (a.x + b.x);
  b.y = 0.5f * (a.y + b.y);
  b.z = 0.5f * (a.z + b.z);
  b.w = 0.5f * (a.w + b.w);
  reinterpret_cast<float4*>(sp)[i] = b;
}

// coeffs[m, j] += sum_{k=0..3} W[m, kbase+k] * B_k[j]   via v_wmma_f32_16x16x4_f32
// A (16x4 f32): VGPR0 = K{0 | 2} across lane halves, VGPR1 = K{1 | 3}.
// B (4x16 f32): VGPR0 = K{0 | 1} rows across lane halves, VGPR1 = K{2 | 3}.
// D rows 0..5 valid (W has 6 rows); lanes 0..15 hold M=0..7 -> only they store.
__global__ void k_wmma_combine(const float* __restrict__ B0, const float* __restrict__ B1,
                               const float* __restrict__ B2, const float* __restrict__ B3,
                               const float* __restrict__ Wm, float* __restrict__ coeffs,
                               long NC, int kbase, int first) {
  int lane = threadIdx.x & 31;
  int wave = blockIdx.x * (blockDim.x >> 5) + (threadIdx.x >> 5);
  long j0 = (long)wave * 16;
  if (j0 >= NC) return;  // uniform per wave
  int half = lane >> 4;
  int m = lane & 15;
  long col = j0 + (lane & 15);

  int ka0 = half ? 2 : 0;
  v2f a;
  a.x = (m < 6 && (kbase + ka0) < 17) ? Wm[m * 17 + kbase + ka0] : 0.f;
  a.y = (m < 6 && (kbase + ka0 + 1) < 17) ? Wm[m * 17 + kbase + ka0 + 1] : 0.f;

  const float* Bx = half ? B1 : B0;
  const float* By = half ? B3 : B2;
  v2f b;
  b.x = Bx[col];
  b.y = By[col];

  v8f c = {};
  if (!first && half == 0) {
#pragma unroll
    for (int r = 0; r < 6; ++r) c[r] = coeffs[(long)r * NC + col];
  }

  v8f d = __builtin_amdgcn_wmma_f32_16x16x4_f32(
      /*neg_a=*/false, a, /*neg_b=*/false, b,
      /*c_mod=*/(short)0, c, /*reuse_a=*/false, /*reuse_b=*/false);

  if (half == 0) {
#pragma unroll
    for (int r = 0; r < 6; ++r) coeffs[(long)r * NC + col] = d[r];
  }
}

// out[n][i][c][{0,1}] = {relu(v), relu(-v)}, v = coeffs[i][n*C+c]. float2 stores.
__global__ void k_expand(const float* __restrict__ coeffs, float* __restrict__ out,
                         long NC, long total) {
  long tid = (long)blockIdx.x * blockDim.x + threadIdx.x;
  if (tid >= total) return;
  long c = tid & (C_FEAT - 1);
  long ni = tid >> 7;       // n*6 + i
  long i = ni % 6;
  long n = ni / 6;
  float v = coeffs[i * NC + n * C_FEAT + c];
  float2 o;
  o.x = v > 0.f ? v : 0.f;
  o.y = v < 0.f ? -v : 0.f;
  reinterpret_cast<float2*>(out)[tid] = o;
}

extern "C" void kernel_launch(void* const* d_in, const int* in_sizes, int n_in,
                              void* d_out, int out_size, void* d_ws, size_t ws_size,
                              hipStream_t stream) {
  const float* x  = (const float*)d_in[0];   // [N, 128] f32
  const int*   ei = (const int*)d_in[1];     // [2, E] i32
  const float* Wm = (const float*)d_in[2];   // [6, 17] f32

  const long NC = (long)in_sizes[0];         // N * 128
  const int  E  = in_sizes[1] / 2;
  const int  N  = (int)(NC / C_FEAT);
  const int* row = ei;
  const int* col = ei + E;

  // Workspace layout (~263 MB): deg | w | 4-slot level ring | coeffs[6, NC]
  char* ws = (char*)d_ws;
  size_t off = 0;
  auto take = [&](size_t bytes) -> char* {
    char* p = ws + off;
    off = (off + bytes + 511) & ~(size_t)511;
    return p;
  };
  float* deg = (float*)take((size_t)N * 4);
  float* w   = (float*)take((size_t)E * 4);
  float* L[4];
  for (int k = 0; k < 4; ++k) L[k] = (float*)take((size_t)NC * 4);
  float* coeffs = (float*)take((size_t)NC * 6 * 4);
  (void)ws_size; (void)n_in; (void)out_size;

  // degree -> deg_inv -> per-edge weight
  hipMemsetAsync(deg, 0, (size_t)N * 4, stream);
  k_deg<<<(E + 255) / 256, 256, 0, stream>>>(col, deg, E);
  k_inv<<<(N + 255) / 256, 256, 0, stream>>>(deg, N);
  k_gatherw<<<(E + 255) / 256, 256, 0, stream>>>(col, deg, w, E);

  const long n4 = NC / 4;
  const int wmma_blocks = (int)((NC / 16 + 7) / 8);  // 8 waves/block, 16 cols/wave

  for (int t = 1; t <= 16; ++t) {
    float* cur = L[t & 3];
    const float* prev = (t == 1) ? x : L[(t - 1) & 3];
    hipMemsetAsync(cur, 0, (size_t)NC * 4, stream);
    k_scatter<<<(E + 7) / 8, 256, 0, stream>>>(row, col, w, prev, cur, E);
    k_combine<<<(int)((n4 + 255) / 256), 256, 0, stream>>>(prev, cur, n4);

    if ((t & 3) == 3) {  // t = 3,7,11,15 -> K-chunks {0..3},{4..7},{8..11},{12..15}
      const float* B0 = (t == 3) ? x : L[0];  // level 0 is the input x
      k_wmma_combine<<<wmma_blocks, 256, 0, stream>>>(
          B0, L[1], L[2], L[3], Wm, coeffs, NC, (t >> 2) * 4, (t == 3) ? 1 : 0);
    }
  }
  // Final K-chunk: level 16 in L[0]; K=17..19 zeroed via the A operand.
  k_wmma_combine<<<wmma_blocks, 256, 0, stream>>>(
      L[0], L[1], L[2], L[3], Wm, coeffs, NC, 16, 0);

  const long total = NC * 6;
  k_expand<<<(int)((total + 255) / 256), 256, 0, stream>>>(coeffs, (float*)d_out, NC, total);
}